// WubuBlock_46342697124315
// MI455X (gfx1250) — compile-verified
//
#include <hip/hip_runtime.h>
#include <math.h>

// ---------------------------------------------------------------------------
// WubuBlock for MI455X (gfx1250, wave32, WMMA).
// fp32 end-to-end on V_WMMA_F32_16X16X4_F32 (matches the fp32 reference).
// GEMMs: 8-wave blocks computing 32x64 macro-tiles; B chunks staged in LDS,
// double-buffered through GLOBAL_LOAD_ASYNC_TO_LDS_B128 (s_wait_asynccnt).
// All GEMM shapes are compile-time so B addressing folds into immediate
// offsets instead of 64-bit VALU mad chains.
// ---------------------------------------------------------------------------

#define EPSV 1e-7f
#define NN 512
#define DD 512
#define HH 8
#define HDIM 64
#define HALFD 32
#define WINW 128
#define DFF 2048
#define NSLOT 144      // 9 key tiles * 16 per 16-query tile (window = 128)
#define SPAD 148
#define KC 32          // GEMM K-chunk staged in LDS

typedef float v2f __attribute__((ext_vector_type(2)));
typedef float v8f __attribute__((ext_vector_type(8)));

#if __has_builtin(__builtin_amdgcn_global_load_async_to_lds_b128)
#define USE_ASYNC 1
typedef int v4i_t __attribute__((vector_size(16)));
typedef __attribute__((address_space(1))) v4i_t* gas_ptr;   // HIP prints as __device__*
typedef __attribute__((address_space(3))) v4i_t* las_ptr;   // HIP prints as __shared__*
#else
#define USE_ASYNC 0
#endif

__device__ __forceinline__ void wait_async_lds() {
#if USE_ASYNC
#if __has_builtin(__builtin_amdgcn_s_wait_asynccnt)
    __builtin_amdgcn_s_wait_asynccnt(0);
#else
    asm volatile("s_wait_asynccnt 0x0" ::: "memory");
#endif
#endif
}

// ------------------------------ LayerNorm ----------------------------------
__global__ __launch_bounds__(256) void ln_kernel(
    const float* __restrict__ x, const float* __restrict__ g,
    const float* __restrict__ b, float* __restrict__ y)
{
    __shared__ float s1[256], s2[256];
    __shared__ float mu_s, rs_s;
    int row = blockIdx.x;
    int t = threadIdx.x;
    float a0 = x[(size_t)row * DD + t];
    float a1 = x[(size_t)row * DD + 256 + t];
    s1[t] = a0 + a1;
    s2[t] = a0 * a0 + a1 * a1;
    __syncthreads();
    for (int off = 128; off; off >>= 1) {
        if (t < off) { s1[t] += s1[t + off]; s2[t] += s2[t + off]; }
        __syncthreads();
    }
    if (t == 0) {
        float mu = s1[0] / (float)DD;
        float var = s2[0] / (float)DD - mu * mu;
        mu_s = mu;
        rs_s = rsqrtf(var + 1e-6f);
    }
    __syncthreads();
    float mu = mu_s, rs = rs_s;
    y[(size_t)row * DD + t]       = (a0 - mu) * rs * g[t] + b[t];
    y[(size_t)row * DD + 256 + t] = (a1 - mu) * rs * g[256 + t] + b[256 + t];
}

// ---- stage one KC x 64 chunk of B into LDS (async path if available) ------
template<int NC>
__device__ __forceinline__ void stage_chunk(const float* __restrict__ Bg,
                                            float* lbuf, int k0, int tn0, int t)
{
    // 2048 floats per chunk, 256 threads, 2x b128 per thread
#pragma unroll
    for (int s = 0; s < 2; ++s) {
        int idx = (t + s * 256) * 4;       // float index within chunk
        int r   = idx >> 6;                // chunk row 0..31
        int col = idx & 63;
        const float* g = Bg + (size_t)(k0 + r) * NC + tn0 + col;
        float* l = lbuf + idx;
#if USE_ASYNC
        __builtin_amdgcn_global_load_async_to_lds_b128(
            (gas_ptr)(size_t)g, (las_ptr)(v4i_t*)l, 0, 0);
#else
        float4 tmp = *(const float4*)g;
        *(float4*)l = tmp;
#endif
    }
}

// ---------------- f32 WMMA GEMM: 8 waves -> 32x64 macro-tile ---------------
// C[512,NC] = act(A[512,K] @ B[K,NC] + bias) (+ resid)
template<int K, int NC, int ACT, int RESID>
__global__ __launch_bounds__(256) void gemm_wmma(
    const float* __restrict__ A, const float* __restrict__ Bm,
    const float* __restrict__ bias, const float* __restrict__ resid,
    float* __restrict__ C)
{
    __shared__ float Bs[2][KC * 64];
    const int t    = threadIdx.x;
    const int wid  = t >> 5;            // 0..7
    const int lane = t & 31;
    const int wm   = wid >> 2;          // 0..1  (M sub-tile)
    const int wn   = wid & 3;           // 0..3  (N sub-tile)
    const int half = lane >> 4;         // K sub-offset select
    const int l15  = lane & 15;

    const int tM  = blockIdx.y * 32 + wm * 16;
    const int tN0 = blockIdx.x * 64;          // macro-tile N base
    const int tN  = tN0 + wn * 16;
    const int bcol = wn * 16 + l15;           // column inside LDS chunk

    const float* Arow = A + (size_t)(tM + l15) * K + (half << 1);

    constexpr int NCHUNK = K / KC;
    v8f acc = {};

    stage_chunk<NC>(Bm, &Bs[0][0], 0, tN0, t);
    for (int c = 0; c < NCHUNK; ++c) {
        wait_async_lds();
        __syncthreads();
        if (c + 1 < NCHUNK)
            stage_chunk<NC>(Bm, &Bs[(c + 1) & 1][0], (c + 1) * KC, tN0, t);
        const float* Bc = &Bs[c & 1][0];
        const float* Ac = Arow + c * KC;
#pragma unroll
        for (int kk = 0; kk < KC; kk += 4) {
            v2f fa = *(const v2f*)(Ac + kk);              // A[m][k..k+1]
            v2f fb;
            fb.x = Bc[(kk + (half << 1)) * 64 + bcol];    // B[k][n] from LDS
            fb.y = Bc[(kk + (half << 1) + 1) * 64 + bcol];
            acc = __builtin_amdgcn_wmma_f32_16x16x4_f32(
                false, fa, false, fb, (short)0, acc, false, false);
        }
    }

#pragma unroll
    for (int r = 0; r < 8; ++r) {
        int row = tM + r + (half << 3);
        int col = tN + l15;
        float val = acc[r] + bias[col];
        if (ACT) {  // tanh GELU
            float u = 0.7978845608028654f * (val + 0.044715f * val * val * val);
            val = 0.5f * val * (1.f + tanhf(u));
        }
        if (RESID) val += resid[(size_t)row * NC + col];
        C[(size_t)row * NC + col] = val;
    }
}

// --------------- Rotary + expmap0 (+ squared-norm tables) ------------------
__global__ __launch_bounds__(256) void rot_exp_kernel(
    const float* __restrict__ qbuf, const float* __restrict__ kbuf,
    const float* __restrict__ freqs, const float* __restrict__ cptr,
    float* __restrict__ qh, float* __restrict__ kh,
    float* __restrict__ qn2, float* __restrict__ kn2)
{
    int wv = (int)((blockIdx.x * blockDim.x + threadIdx.x) >> 5);
    int lane = threadIdx.x & 31;
    int src = wv >> 12;            // 0 = q, 1 = k (4096 jobs each)
    int idx = wv & 4095;
    int n = idx >> 3;
    int h = idx & 7;
    const float* in  = src ? kbuf : qbuf;
    float* outp      = src ? kh   : qh;
    float* n2p       = src ? kn2  : qn2;

    float xr = in[(size_t)n * DD + h * HDIM + lane];
    float xi = in[(size_t)n * DD + h * HDIM + 32 + lane];
    float fc = freqs[(size_t)(n * HALFD + lane) * 2 + 0];  // cos
    float fs = freqs[(size_t)(n * HALFD + lane) * 2 + 1];  // sin
    float rr = xr * fc - xi * fs;
    float ri = xr * fs + xi * fc;

    float p = rr * rr + ri * ri;
    for (int off = 16; off; off >>= 1) p += __shfl_xor(p, off, 32);
    float vn  = sqrtf(p);
    float cc  = cptr[0];
    float sqc = fmaxf(sqrtf(cc), EPSV);
    float svn = fmaxf(vn, EPSV);
    float mag = tanhf(sqc * svn) / sqc;
    bool  zer = vn < EPSV;
    float scl = zer ? 0.f : (mag / svn);

    outp[((size_t)h * NN + n) * HDIM + lane]      = scl * rr;
    outp[((size_t)h * NN + n) * HDIM + 32 + lane] = scl * ri;
    if (lane == 0) n2p[h * NN + n] = zer ? 0.f : mag * mag;
}

// --------------------- Windowed Poincare attention -------------------------
__global__ __launch_bounds__(128) void attn_kernel(
    const float* __restrict__ qh, const float* __restrict__ kh,
    const float* __restrict__ qn2, const float* __restrict__ kn2,
    const float* __restrict__ vbuf, const float* __restrict__ cptr,
    const float* __restrict__ gsp, float* __restrict__ attn)
{
    __shared__ float sw[4][16][SPAD];
    int w = threadIdx.x >> 5;
    int lane = threadIdx.x & 31;
    int wid = blockIdx.x * 4 + w;
    int h  = wid >> 5;
    int qt = wid & 31;
    int n0 = qt << 4;
    int mbase = (qt - 8) << 4;        // tile-aligned; may be negative
    int half = lane >> 4, l15 = lane & 15;

    float cc  = cptr[0];
    float sqc = fmaxf(sqrtf(cc), EPSV);
    float gs  = gsp[h];

    const float* qrow = qh + ((size_t)h * NN + n0 + l15) * HDIM + (half << 1);

    // ---- scores: S = Qh . Kh^T, then Poincare distance epilogue ----
    for (int kt = 0; kt < 9; ++kt) {
        int mt = mbase + kt * 16;
        if (mt < 0) {
            for (int r = 0; r < 8; ++r)
                sw[w][r + (half << 3)][kt * 16 + l15] = -1e30f;
            continue;
        }
        v8f acc = {};
        const float* krow = kh + ((size_t)h * NN + mt + l15) * HDIM + (half << 1);
#pragma unroll
        for (int k = 0; k < HDIM; k += 4) {
            v2f fa = *(const v2f*)(qrow + k);
            v2f fb = *(const v2f*)(krow + k);
            acc = __builtin_amdgcn_wmma_f32_16x16x4_f32(
                false, fa, false, fb, (short)0, acc, false, false);
        }
#pragma unroll
        for (int r = 0; r < 8; ++r) {
            int qr = r + (half << 3);
            int n  = n0 + qr;
            int m  = mt + l15;
            float xy = acc[r];
            float x2 = qn2[h * NN + n];
            float y2 = kn2[h * NN + m];
            // ||a*x - b*y||^2 = a^2 x2 - 2ab xy + b^2 y2
            float A_  = 1.f - 2.f * cc * xy + cc * y2;
            float B_  = 1.f - cc * x2;
            float den = fmaxf(1.f - 2.f * cc * xy + cc * cc * x2 * y2, EPSV);
            float nm2 = fmaxf(A_ * A_ * x2 - 2.f * A_ * B_ * xy + B_ * B_ * y2, 0.f);
            float dn  = sqrtf(nm2) / den;
            float arg = fminf(sqc * fmaxf(dn, EPSV), 1.f - EPSV);
            float dist = 2.f * atanhf(arg) / sqc;
            bool valid = (m <= n) && ((n - m) < WINW);
            sw[w][qr][kt * 16 + l15] = valid ? (-gs * dist) : -1e30f;
        }
    }

    // ---- softmax per query row (same-wave LDS ordering, no barrier) ----
    if (lane < 16) {
        float mx = -1e30f;
        for (int s = 0; s < NSLOT; ++s) mx = fmaxf(mx, sw[w][lane][s]);
        float sum = 0.f;
        for (int s = 0; s < NSLOT; ++s) {
            float e = __expf(sw[w][lane][s] - mx);
            sw[w][lane][s] = e;
            sum += e;
        }
        float inv = 1.f / sum;
        for (int s = 0; s < NSLOT; ++s) sw[w][lane][s] *= inv;
    }

    // ---- attn = W @ V (masked slots carry weight 0) ----
    for (int ct = 0; ct < 4; ++ct) {
        int c0 = ct << 4;
        v8f acc = {};
        const float* vcol = vbuf + h * HDIM + c0 + l15;
#pragma unroll
        for (int k = 0; k < NSLOT; k += 4) {
            int kk = k + (half << 1);
            v2f fa;
            fa.x = sw[w][l15][kk];
            fa.y = sw[w][l15][kk + 1];
            int m0 = mbase + kk;
            int mA = m0 < 0 ? 0 : m0;
            int mB = (m0 + 1) < 0 ? 0 : (m0 + 1);
            v2f fb;
            fb.x = vcol[(size_t)mA * DD];
            fb.y = vcol[(size_t)mB * DD];
            acc = __builtin_amdgcn_wmma_f32_16x16x4_f32(
                false, fa, false, fb, (short)0, acc, false, false);
        }
#pragma unroll
        for (int r = 0; r < 8; ++r) {
            int n = n0 + r + (half << 3);
            attn[(size_t)n * DD + h * HDIM + c0 + l15] = acc[r];
        }
    }
}

// ---------------------------------------------------------------------------
extern "C" void kernel_launch(void* const* d_in, const int* in_sizes, int n_in,
                              void* d_out, int out_size, void* d_ws, size_t ws_size,
                              hipStream_t stream) {
    const float* x    = (const float*)d_in[0];
    const float* c    = (const float*)d_in[1];
    const float* Wq   = (const float*)d_in[2];
    const float* bq   = (const float*)d_in[3];
    const float* Wk   = (const float*)d_in[4];
    const float* bk   = (const float*)d_in[5];
    const float* Wv   = (const float*)d_in[6];
    const float* bv   = (const float*)d_in[7];
    const float* Wo   = (const float*)d_in[8];
    const float* bo   = (const float*)d_in[9];
    const float* W1   = (const float*)d_in[10];
    const float* b1   = (const float*)d_in[11];
    const float* W2   = (const float*)d_in[12];
    const float* b2   = (const float*)d_in[13];
    const float* ln1g = (const float*)d_in[14];
    const float* ln1b = (const float*)d_in[15];
    const float* ln2g = (const float*)d_in[16];
    const float* ln2b = (const float*)d_in[17];
    const float* gs   = (const float*)d_in[18];
    const float* freqs = (const float*)d_in[19];   // complex64 -> interleaved f32
    float* out = (float*)d_out;

    const size_t NT = (size_t)NN * DD;   // 262144
    float* ws   = (float*)d_ws;
    float* xn    = ws;
    float* qbuf  = xn   + NT;
    float* kbuf  = qbuf + NT;
    float* vbuf  = kbuf + NT;
    float* qhb   = vbuf + NT;
    float* khb   = qhb  + NT;
    float* x1    = khb  + NT;
    float* hb    = x1   + NT;
    float* attnb = hb   + NT;
    float* qn2   = attnb + NT;           // H*N = 4096
    float* kn2   = qn2  + (size_t)HH * NN;
    float* ffn1  = kn2  + (size_t)HH * NN;  // 512*2048

    // 1) LN1
    ln_kernel<<<NN, 256, 0, stream>>>(x, ln1g, ln1b, xn);
    // 2) QKV projections (f32 WMMA, LDS-staged B)
    dim3 g512(DD / 64, NN / 32);     // (8,16)
    gemm_wmma<512, 512, 0, 0><<<g512, 256, 0, stream>>>(xn, Wq, bq, nullptr, qbuf);
    gemm_wmma<512, 512, 0, 0><<<g512, 256, 0, stream>>>(xn, Wk, bk, nullptr, kbuf);
    gemm_wmma<512, 512, 0, 0><<<g512, 256, 0, stream>>>(xn, Wv, bv, nullptr, vbuf);
    // 3) rotary + expmap0 + norm tables (2*N*H waves)
    rot_exp_kernel<<<1024, 256, 0, stream>>>(qbuf, kbuf, freqs, c,
                                             qhb, khb, qn2, kn2);
    // 4) windowed Poincare attention (8 heads * 32 query tiles, 4 waves/block)
    attn_kernel<<<64, 128, 0, stream>>>(qhb, khb, qn2, kn2, vbuf, c, gs, attnb);
    // 5) output projection + residual
    gemm_wmma<512, 512, 0, 1><<<g512, 256, 0, stream>>>(attnb, Wo, bo, x, x1);
    // 6) LN2
    ln_kernel<<<NN, 256, 0, stream>>>(x1, ln2g, ln2b, hb);
    // 7) MLP up + GELU
    dim3 gup(DFF / 64, NN / 32);     // (32,16)
    gemm_wmma<512, 2048, 1, 0><<<gup, 256, 0, stream>>>(hb, W1, b1, nullptr, ffn1);
    // 8) MLP down + residual -> out
    gemm_wmma<2048, 512, 0, 1><<<g512, 256, 0, stream>>>(ffn1, W2, b2, x1, out);
}